// SAGE_12841952215815
// MI455X (gfx1250) — compile-verified
//
#include <hip/hip_runtime.h>
#include <hip/hip_bf16.h>

#define N_NODES 100000
#define N_EDGES 1600000
#define HID 128
#define OUTD 16
#define N_GRAPHS 128

typedef __attribute__((ext_vector_type(2))) float v2f;
typedef __attribute__((ext_vector_type(8))) float v8f;

// ---------------------------------------------------------------------------
// Degree: deg[dst[e]] += 1
// ---------------------------------------------------------------------------
__global__ void deg_kernel(const int* __restrict__ dst, float* __restrict__ deg,
                           int n_edges) {
    int e = blockIdx.x * 256 + threadIdx.x;
    if (e >= n_edges) return;
    unsafeAtomicAdd(&deg[dst[e]], 1.0f);
}

// ---------------------------------------------------------------------------
// Edge aggregation: agg[dst[e]][:] += h[src[e]][:]
// One wave (32 lanes) per edge; each lane moves a float4 (coalesced 512B row).
// ---------------------------------------------------------------------------
__global__ void agg_kernel(const float* __restrict__ h,
                           const int* __restrict__ src,
                           const int* __restrict__ dst,
                           float* __restrict__ agg, int n_edges) {
    unsigned tid = blockIdx.x * 256u + threadIdx.x;
    int e = tid >> 5;
    int lane = tid & 31;
    if (e >= n_edges) return;
    int s = src[e];
    int d = dst[e];
    const float4 v = *(const float4*)(h + (size_t)s * HID + lane * 4);
    float* base = agg + (size_t)d * HID + lane * 4;
    unsafeAtomicAdd(base + 0, v.x);
    unsafeAtomicAdd(base + 1, v.y);
    unsafeAtomicAdd(base + 2, v.z);
    unsafeAtomicAdd(base + 3, v.w);
}

// ---------------------------------------------------------------------------
// Fused SAGE layer: out = maybe_relu( ((agg + h) / (deg+1)) @ W + b )
// 256 threads = 8 waves; each wave computes a 16x128 output tile with
// V_WMMA_F32_16X16X4_F32 (exact fp32).
//
// Fragment layouts per CDNA5 ISA 7.12.2 (wave32):
//   A 16x4 :  M = lane&15,  K = vgpr + 2*(lane>=16)
//   B 4x16 :  N = lane&15,  K = vgpr + 2*(lane>=16)
//   C 16x16:  N = lane&15,  M = vgpr + 8*(lane>=16)
//
// W is staged in LDS K-pair interleaved:
//   sW[kpair*SWLD + 2*n + j] = W[2*kpair + j][n]   (j = 0,1)
// so a B fragment is ONE contiguous ds_load_b64 into an even VGPR pair
// (no cross-register shuffles). Row stride SWLD = 288 floats == 32 banks
// (mod 64): lanes 0-15 (kpair p) and lanes 16-31 (kpair p+1) hit disjoint
// 32-bank halves -> conflict-free.
// ---------------------------------------------------------------------------
#define SWLD 288  // floats per k-pair row (128 float2 + 16 float2 pad)

__global__ __launch_bounds__(256) void layer_kernel(
    const float* __restrict__ h_in, const float* __restrict__ agg,
    const float* __restrict__ deg, const float* __restrict__ W,
    const float* __restrict__ bias, float* __restrict__ out, int relu,
    int n_nodes) {
    __shared__ float sW[(HID / 2) * SWLD];  // 64 * 288 * 4B = 72 KB

    // Cooperative staging: coalesced global reads, scattered LDS writes.
    for (int idx = threadIdx.x; idx < HID * HID; idx += 256) {
        int k = idx >> 7;   // row of W
        int c = idx & 127;  // col of W
        sW[(k >> 1) * SWLD + 2 * c + (k & 1)] = W[idx];
    }
    __syncthreads();

    const int wave = threadIdx.x >> 5;  // 0..7
    const int lane = threadIdx.x & 31;
    const int m = lane & 15;
    const int half = lane >> 4;  // 0 or 1
    const int r0 = blockIdx.x * 128 + wave * 16;

    // A-fragment source row for this lane (select-to-zero for tail: EXEC
    // must stay all-1s for WMMA, so no divergent branch here).
    const int row = r0 + m;
    const bool valid = row < n_nodes;
    const int rowc = valid ? row : 0;
    const float rdeg = valid ? 1.0f / (deg[rowc] + 1.0f) : 0.0f;
    const float2* hrow = (const float2*)(h_in + (size_t)rowc * HID);
    const float2* arow = (const float2*)(agg + (size_t)rowc * HID);

    v8f acc[8] = {};  // 8 column tiles -> 16x128 output

#pragma unroll 4
    for (int kk = 0; kk < HID; kk += 4) {
        // This lane's K base: kb = kk + 2*half  ->  k-pair index:
        const int kpair = (kk >> 1) + half;
        const float2 hv = hrow[kpair];
        const float2 av = arow[kpair];
        v2f a;
        a.x = (hv.x + av.x) * rdeg;
        a.y = (hv.y + av.y) * rdeg;
        const float* brow = sW + kpair * SWLD;
#pragma unroll
        for (int nt = 0; nt < 8; ++nt) {
            const int ncol = nt * 16 + m;
            const float2 bf = *(const float2*)(brow + 2 * ncol);
            v2f b;
            b.x = bf.x;  // W[kb][ncol]
            b.y = bf.y;  // W[kb+1][ncol]
            acc[nt] = __builtin_amdgcn_wmma_f32_16x16x4_f32(
                false, a, false, b, (short)0, acc[nt], false, false);
        }
    }

    // Epilogue: bias (+ReLU), bounds-guarded store.
#pragma unroll
    for (int nt = 0; nt < 8; ++nt) {
        const int ncol = nt * 16 + m;
        const float bv = bias[ncol];
#pragma unroll
        for (int v = 0; v < 8; ++v) {
            const int orow = r0 + v + 8 * half;
            if (orow < n_nodes) {
                float val = acc[nt][v] + bv;
                if (relu) val = fmaxf(val, 0.0f);
                out[(size_t)orow * HID + ncol] = val;
            }
        }
    }
}

// ---------------------------------------------------------------------------
// Gate + weighted per-graph accumulation.
// One wave per node: gate = sigmoid(h . Wv + bv);
// num[g][:] += gate*h[:]; den[g] += gate.
// ---------------------------------------------------------------------------
__global__ void gate_kernel(const float* __restrict__ h,
                            const int* __restrict__ graph_id,
                            const float* __restrict__ Wv,
                            const float* __restrict__ bv,
                            float* __restrict__ num, float* __restrict__ den,
                            int n_nodes) {
    unsigned tid = blockIdx.x * 256u + threadIdx.x;
    int node = tid >> 5;
    int lane = tid & 31;
    if (node >= n_nodes) return;
    const float4 hv = *(const float4*)(h + (size_t)node * HID + lane * 4);
    const float4 wv = *(const float4*)(Wv + lane * 4);
    float d = hv.x * wv.x + hv.y * wv.y + hv.z * wv.z + hv.w * wv.w;
#pragma unroll
    for (int o = 16; o >= 1; o >>= 1) d += __shfl_xor(d, o, 32);
    const float g = 1.0f / (1.0f + __expf(-(d + bv[0])));
    const int gid = graph_id[node];
    float* nb = num + (size_t)gid * HID + lane * 4;
    unsafeAtomicAdd(nb + 0, g * hv.x);
    unsafeAtomicAdd(nb + 1, g * hv.y);
    unsafeAtomicAdd(nb + 2, g * hv.z);
    unsafeAtomicAdd(nb + 3, g * hv.w);
    if (lane == 0) unsafeAtomicAdd(&den[gid], g);
}

// ---------------------------------------------------------------------------
// Final readout: out[g][o] = bc[o] + sum_k (num[g][k]/den[g]) * Wc[k][o]
// 128*16 = 2048 threads, k-loop of 128. Trivial cost.
// ---------------------------------------------------------------------------
__global__ void readout_kernel(const float* __restrict__ num,
                               const float* __restrict__ den,
                               const float* __restrict__ Wc,
                               const float* __restrict__ bc,
                               float* __restrict__ out) {
    int t = blockIdx.x * 256 + threadIdx.x;
    if (t >= N_GRAPHS * OUTD) return;
    int g = t >> 4;
    int o = t & 15;
    const float invd = 1.0f / den[g];
    float s = bc[o];
#pragma unroll 4
    for (int k = 0; k < HID; ++k)
        s += num[g * HID + k] * invd * Wc[k * OUTD + o];
    out[t] = s;
}

// ---------------------------------------------------------------------------
extern "C" void kernel_launch(void* const* d_in, const int* in_sizes, int n_in,
                              void* d_out, int out_size, void* d_ws,
                              size_t ws_size, hipStream_t stream) {
    const float* x = (const float*)d_in[0];
    const int* src = (const int*)d_in[1];
    const int* dst = (const int*)d_in[2];
    const int* gid = (const int*)d_in[3];
    const float* W1 = (const float*)d_in[4];
    const float* b1 = (const float*)d_in[5];
    const float* W2 = (const float*)d_in[6];
    const float* b2 = (const float*)d_in[7];
    const float* W3 = (const float*)d_in[8];
    const float* b3 = (const float*)d_in[9];
    const float* Wv = (const float*)d_in[10];
    const float* bv = (const float*)d_in[11];
    const float* Wc = (const float*)d_in[12];
    const float* bc = (const float*)d_in[13];

    const size_t NF = (size_t)N_NODES * HID;
    float* ws = (float*)d_ws;
    float* agg = ws;             // [N, 128]
    float* hA = ws + NF;         // [N, 128]
    float* hB = ws + 2 * NF;     // [N, 128]
    float* deg = ws + 3 * NF;    // [N]
    float* num = deg + N_NODES;  // [G, 128]
    float* den = num + (size_t)N_GRAPHS * HID;  // [G]

    const int edgeBlocks = (N_EDGES + 255) / 256;
    const int aggBlocks = (int)(((size_t)N_EDGES * 32 + 255) / 256);
    const int layerBlocks = (N_NODES + 127) / 128;
    const int gateBlocks = (int)(((size_t)N_NODES * 32 + 255) / 256);

    hipMemsetAsync(deg, 0, N_NODES * sizeof(float), stream);
    hipMemsetAsync(num, 0, (size_t)(N_GRAPHS * HID + N_GRAPHS) * sizeof(float),
                   stream);

    deg_kernel<<<edgeBlocks, 256, 0, stream>>>(dst, deg, N_EDGES);

    // Layer 1: x -> hA (ReLU)
    hipMemsetAsync(agg, 0, NF * sizeof(float), stream);
    agg_kernel<<<aggBlocks, 256, 0, stream>>>(x, src, dst, agg, N_EDGES);
    layer_kernel<<<layerBlocks, 256, 0, stream>>>(x, agg, deg, W1, b1, hA, 1,
                                                  N_NODES);

    // Layer 2: hA -> hB (ReLU)
    hipMemsetAsync(agg, 0, NF * sizeof(float), stream);
    agg_kernel<<<aggBlocks, 256, 0, stream>>>(hA, src, dst, agg, N_EDGES);
    layer_kernel<<<layerBlocks, 256, 0, stream>>>(hA, agg, deg, W2, b2, hB, 1,
                                                  N_NODES);

    // Layer 3: hB -> hA (no ReLU)
    hipMemsetAsync(agg, 0, NF * sizeof(float), stream);
    agg_kernel<<<aggBlocks, 256, 0, stream>>>(hB, src, dst, agg, N_EDGES);
    layer_kernel<<<layerBlocks, 256, 0, stream>>>(hB, agg, deg, W3, b3, hA, 0,
                                                  N_NODES);

    // Gated weighted-mean readout + final projection
    gate_kernel<<<gateBlocks, 256, 0, stream>>>(hA, gid, Wv, bv, num, den,
                                                N_NODES);
    readout_kernel<<<8, 256, 0, stream>>>(num, den, Wc, bc, (float*)d_out);
}